// Attention_layer_68942815035578
// MI455X (gfx1250) — compile-verified
//
#include <hip/hip_runtime.h>
#include <hip/hip_bf16.h>

// Problem constants (from reference)
constexpr int BATCH   = 2;
constexpr int SEQ     = 2048;
constexpr int EMB     = 128;
constexpr int NHEADS  = 8;
constexpr int HDIM    = 16;        // EMB / NHEADS
constexpr int KTILES  = SEQ / 16;  // 128

typedef __attribute__((ext_vector_type(2))) float v2f;
typedef __attribute__((ext_vector_type(8))) float v8f;

// -------------------------------------------------------------------------
// Kernel 1: QKV projection.  One wave computes one 16x16 tile of X @ W via
// 32 chained V_WMMA_F32_16X16X4_F32 ops (K = 128), then scatters the tile
// into per-head (B, H, S, 16) layout.  Head split per reference:
//   column e of the projection maps to head h = e % 8, dim d = e / 8.
// WMMA fp32 operand layout (wave32):
//   A (16x4):  lane m = lane&15, a.x = A[m][kb], a.y = A[m][kb+1], kb = (lane>>4)*2
//   B (4x16):  lane n = lane&15, b.x = B[kb][n], b.y = B[kb+1][n]
//   C (16x16): vgpr r holds row (r + 8*(lane>>4)), col = lane&15
// -------------------------------------------------------------------------
__global__ __launch_bounds__(32) void qkv_proj_kernel(
    const float* __restrict__ X,    // (B,S,EMB) flattened rows = B*S
    const float* __restrict__ Wq,
    const float* __restrict__ Wk,
    const float* __restrict__ Wv,
    float* __restrict__ Qws,        // (B,H,S,HDIM)
    float* __restrict__ Kws,
    float* __restrict__ Vws)
{
    const int lane = threadIdx.x;          // 0..31, full wave (EXEC all ones)
    const int mt   = blockIdx.x;           // 0..255 : 16-row tile over B*S
    const int z    = blockIdx.y;           // 0..23  : which-matrix * 8 + n-tile
    const int which = z >> 3;
    const int nt    = z & 7;

    const float* W = (which == 0) ? Wq : (which == 1) ? Wk : Wv;
    float*       O = (which == 0) ? Qws : (which == 1) ? Kws : Vws;

    const int m    = lane & 15;
    const int half = lane >> 4;
    const int kb   = half * 2;
    const int n    = lane & 15;
    const int n0   = nt * 16;

    const float* xrow = X + (size_t)(mt * 16 + m) * EMB;

    v8f c = {};
    #pragma unroll
    for (int k0 = 0; k0 < EMB; k0 += 4) {
        v2f a, b;
        a.x = xrow[k0 + kb];
        a.y = xrow[k0 + kb + 1];
        b.x = W[(size_t)(k0 + kb)     * EMB + n0 + n];
        b.y = W[(size_t)(k0 + kb + 1) * EMB + n0 + n];
        c = __builtin_amdgcn_wmma_f32_16x16x4_f32(false, a, false, b,
                                                  (short)0, c, false, false);
    }

    // Scatter: column e -> (h, d); row -> (b, s)
    const int e = n0 + n;
    const int h = e & 7;
    const int d = e >> 3;
    #pragma unroll
    for (int r = 0; r < 8; ++r) {
        const int grow  = mt * 16 + r + half * 8;   // flat b*SEQ + s
        const int bidx  = grow >> 11;               // / 2048
        const int s     = grow & (SEQ - 1);
        O[(((size_t)bidx * NHEADS + h) * SEQ + s) * HDIM + d] = c[r];
    }
}

// -------------------------------------------------------------------------
// Kernel 2: flash-style attention, fully TRANSPOSED formulation.
//   S^T = K * Q^T   (keys are C-layout rows, queries are lanes)
//   O^T = V^T * P^T (dims are C-layout rows, queries are lanes)
// Softmax over keys is then lane-local: 7 VALU ops across the 8 score VGPRs
// plus one __shfl_xor(16) to merge halves; running (max, sum) are one scalar
// per lane; alpha rescales acc directly (queries = lanes in both layouts).
// Key mask handled as a single __ballot bitmask per tile.
// -------------------------------------------------------------------------
__global__ __launch_bounds__(32) void attn_kernel(
    const float* __restrict__ Qws,
    const float* __restrict__ Kws,
    const float* __restrict__ Vws,
    const int*   __restrict__ mask_src,   // (B,S)
    const int*   __restrict__ masked_flag,
    float*       __restrict__ out)        // (B,S,EMB)
{
    __shared__ float lds[16][17];         // P^T tile staging (padded)

    const int lane = threadIdx.x;
    const int bid  = blockIdx.x;          // b*H*128 + h*128 + qt
    const int qt   = bid & (KTILES - 1);
    const int h    = (bid >> 7) & (NHEADS - 1);
    const int b    = bid >> 10;

    const float* Qh = Qws + ((size_t)b * NHEADS + h) * SEQ * HDIM;
    const float* Kh = Kws + ((size_t)b * NHEADS + h) * SEQ * HDIM;
    const float* Vh = Vws + ((size_t)b * NHEADS + h) * SEQ * HDIM;
    const int*   mb = mask_src + (size_t)b * SEQ;
    const int use_mask = (*masked_flag == 0);   // encoder path applies key mask

    const int m    = lane & 15;   // A-row index (key for S^T, dim for O^T)
    const int half = lane >> 4;
    const int kb   = half * 2;
    const int n    = lane & 15;   // B/C column index (query)

    // Q^T tile as B-operand fragments: B[d][n] = Q[qt*16 + n][d]
    v2f qb[4];
    #pragma unroll
    for (int c4 = 0; c4 < 4; ++c4) {
        const int d0 = c4 * 4 + kb;
        qb[c4].x = Qh[(size_t)(qt * 16 + n) * HDIM + d0];
        qb[c4].y = Qh[(size_t)(qt * 16 + n) * HDIM + d0 + 1];
    }

    v8f acc = {};                 // O^T: vgpr r = dim (r + 8*half), lane = query
    float run_m = -1e30f;         // per-query running max (dup across halves)
    float run_l = 0.0f;           // per-query running sum

    for (int kt = 0; kt < KTILES; ++kt) {
        if (kt + 1 < KTILES) {    // hint next K/V tiles toward the caches
            __builtin_prefetch(&Kh[(size_t)(kt + 1) * 16 * HDIM], 0, 3);
            __builtin_prefetch(&Vh[(size_t)(kt + 1) * 16 * HDIM], 0, 3);
        }

        // S^T tile = K * Q^T : A[key][d] = K[kt*16 + key][d]
        v8f sc = {};
        #pragma unroll
        for (int c4 = 0; c4 < 4; ++c4) {
            const int d0 = c4 * 4 + kb;
            v2f ka;
            ka.x = Kh[(size_t)(kt * 16 + m) * HDIM + d0];
            ka.y = Kh[(size_t)(kt * 16 + m) * HDIM + d0 + 1];
            sc = __builtin_amdgcn_wmma_f32_16x16x4_f32(false, ka, false, qb[c4],
                                                       (short)0, sc, false, false);
        }

        // Key mask as a wave-wide bitmask: bit j set => key (kt*16+j) masked.
        // Lanes 0-15 and 16-31 load the same 16 keys, bits 0..15 are valid.
        const unsigned bal =
            (unsigned)__ballot(use_mask && (mb[kt * 16 + (lane & 15)] == 0));

        // Scale + mask + per-query (lane-local) max over this tile's 16 keys
        float s[8];
        float tmax = -1e30f;
        #pragma unroll
        for (int r = 0; r < 8; ++r) {
            const int j = r + half * 8;                 // key index in tile
            const float madd = ((bal >> j) & 1u) ? -1000.0f : 0.0f;
            s[r] = sc[r] * 0.25f + madd;                // 1/sqrt(16)
            tmax = fmaxf(tmax, s[r]);
        }
        tmax = fmaxf(tmax, __shfl_xor(tmax, 16, 32));   // merge halves

        const float mnew  = fmaxf(run_m, tmax);
        const float alpha = __expf(run_m - mnew);
        float ps = 0.0f;
        #pragma unroll
        for (int r = 0; r < 8; ++r) {
            const float p = __expf(s[r] - mnew);
            sc[r] = p;                                   // P^T stays in C layout
            ps += p;
        }
        ps += __shfl_xor(ps, 16, 32);                    // merge halves
        run_l = run_l * alpha + ps;
        run_m = mnew;
        #pragma unroll
        for (int r = 0; r < 8; ++r) acc[r] *= alpha;     // per-query rescale

        // P^T (C layout, key-major) -> LDS so it can be reloaded as B operand
        #pragma unroll
        for (int r = 0; r < 8; ++r) lds[r + half * 8][n] = sc[r];
        __syncthreads();

        // acc += V^T * P^T : A[dim][kk] = V[kt*16 + kk][dim], B from LDS
        #pragma unroll
        for (int c4 = 0; c4 < 4; ++c4) {
            const int kk0 = c4 * 4 + kb;
            v2f va, pb;
            va.x = Vh[(size_t)(kt * 16 + kk0)     * HDIM + m];
            va.y = Vh[(size_t)(kt * 16 + kk0 + 1) * HDIM + m];
            pb.x = lds[kk0][n];
            pb.y = lds[kk0 + 1][n];
            acc = __builtin_amdgcn_wmma_f32_16x16x4_f32(false, va, false, pb,
                                                        (short)0, acc, false, false);
        }
        __syncthreads();
    }

    // O^T element (dim = r + 8*half, query = n):
    // out[b][qt*16 + n][dim*8 + h] = acc / run_l  (run_l is per-query = per-lane)
    const float inv_l = 1.0f / run_l;
    #pragma unroll
    for (int r = 0; r < 8; ++r) {
        const int d = r + half * 8;
        out[((size_t)b * SEQ + (qt * 16 + n)) * EMB + d * NHEADS + h] =
            acc[r] * inv_l;
    }
}

// -------------------------------------------------------------------------
extern "C" void kernel_launch(void* const* d_in, const int* in_sizes, int n_in,
                              void* d_out, int out_size, void* d_ws, size_t ws_size,
                              hipStream_t stream) {
    const float* X        = (const float*)d_in[0];
    const int*   mask_src = (const int*)  d_in[1];
    // d_in[2] = target mask (all ones, unused by either path's math)
    const int*   masked   = (const int*)  d_in[3];
    const float* Wq       = (const float*)d_in[4];
    const float* Wk       = (const float*)d_in[5];
    const float* Wv       = (const float*)d_in[6];
    float*       out      = (float*)d_out;

    const size_t head_elems = (size_t)BATCH * NHEADS * SEQ * HDIM; // 524288
    float* Qws = (float*)d_ws;
    float* Kws = Qws + head_elems;
    float* Vws = Kws + head_elems;

    // Projection: 256 row-tiles x (3 matrices * 8 col-tiles), 1 wave each
    dim3 gproj(BATCH * SEQ / 16, 3 * (EMB / 16));
    qkv_proj_kernel<<<gproj, 32, 0, stream>>>(X, Wq, Wk, Wv, Qws, Kws, Vws);

    // Attention: one wave per (b, h, 16-query tile) = 2*8*128 = 2048 waves
    attn_kernel<<<BATCH * NHEADS * KTILES, 32, 0, stream>>>(
        Qws, Kws, Vws, mask_src, masked, out);
}